// GIN_5119601017052
// MI455X (gfx1250) — compile-verified
//
#include <hip/hip_runtime.h>

typedef float v2f __attribute__((ext_vector_type(2)));
typedef float v8f __attribute__((ext_vector_type(8)));

#define HIDF 128
#define NB_RED 1024
#define LN_EPSF 1e-5f

// ---------------- zero buffer ----------------
__global__ void gin_zero(float* __restrict__ p, int M) {
    int i = blockIdx.x * blockDim.x + threadIdx.x;
    if (i < M) p[i] = 0.0f;
}

// ---------------- scatter add: agg[dst] += h[src] ----------------
// one thread per (edge, 4-feature group): E*32 threads
__global__ void gin_scatter(const float* __restrict__ h, const int* __restrict__ ei,
                            float* __restrict__ agg, int E) {
    long long tid = (long long)blockIdx.x * blockDim.x + threadIdx.x;
    int e = (int)(tid >> 5);
    int f = ((int)tid & 31) * 4;
    if (e < E) {
        int s = ei[e];
        int d = ei[E + e];
        const float4 v = *(const float4*)(h + (size_t)s * HIDF + f);
        float* p = agg + (size_t)d * HIDF + f;
        atomicAdd(p + 0, v.x);
        atomicAdd(p + 1, v.y);
        atomicAdd(p + 2, v.z);
        atomicAdd(p + 3, v.w);
    }
}

// ---------------- fused 128-wide GEMM via V_WMMA_F32_16X16X4_F32 ----------------
// out[rows_pad,128] = act( (A (+ Aadd)) @ W + bias ),  W row-major [128][128]
// block = 128 threads (4 waves), each wave computes a 16x128 output tile.
// `out` MUST be padded to a multiple of 64 rows (stores are unguarded);
// A/Aadd reads are clamped to n_rows.
template <bool RELU, bool ADD>
__global__ __launch_bounds__(128) void gin_gemm128(
    const float* __restrict__ A, const float* __restrict__ Aadd,
    const float* __restrict__ W, const float* __restrict__ bias,
    float* __restrict__ out, int n_rows) {
    __shared__ float At[4][16][132];  // +4 pad: avoid 16-way bank conflicts

    const int tid  = threadIdx.x;
    const int wave = tid >> 5;
    const int lane = tid & 31;
    const int rowBase = blockIdx.x * 64 + wave * 16;

    // stage A tile (x + agg) into LDS, clamped rows for the tail block
    for (int idx = lane; idx < 16 * HIDF; idx += 32) {
        int r = idx >> 7, c = idx & 127;
        int gr = rowBase + r;
        gr = gr < n_rows ? gr : n_rows - 1;
        float v = A[(size_t)gr * HIDF + c];
        if (ADD) v += Aadd[(size_t)gr * HIDF + c];
        At[wave][r][c] = v;
    }
    // wave-private LDS slice; compiler inserts s_wait_dscnt before reads

    const int m    = lane & 15;
    const int ksel = (lane < 16) ? 0 : 2;

    v8f acc[8];
#pragma unroll
    for (int j = 0; j < 8; ++j) {
        float bv = bias[j * 16 + m];
        v8f t;
#pragma unroll
        for (int q = 0; q < 8; ++q) t[q] = bv;
        acc[j] = t;
    }

#pragma unroll 2
    for (int kk = 0; kk < 32; ++kk) {
        const int k = kk * 4 + ksel;
        v2f a;
        a.x = At[wave][m][k];
        a.y = At[wave][m][k + 1];
        const float* wp = W + (size_t)k * HIDF + m;
#pragma unroll
        for (int j = 0; j < 8; ++j) {
            v2f b;
            b.x = wp[j * 16];
            b.y = wp[HIDF + j * 16];
            acc[j] = __builtin_amdgcn_wmma_f32_16x16x4_f32(
                false, a, false, b, (short)0, acc[j], false, false);
        }
    }

    // C/D layout: VGPR r -> M=r (lanes 0-15), M=r+8 (lanes 16-31); N = lane%16
    // single base pointer + compile-time immediate offsets, no predication
    const int rofs = (lane < 16) ? 0 : 8;
    float* op = out + (size_t)(rowBase + rofs) * HIDF + m;
#pragma unroll
    for (int j = 0; j < 8; ++j) {
        v8f v = acc[j];
#pragma unroll
        for (int r = 0; r < 8; ++r) {
            float x = v[r];
            if (RELU) x = fmaxf(x, 0.0f);
            op[r * HIDF + j * 16] = x;
        }
    }
}

// ---------------- LayerNorm (graph mode): deterministic tree reduction ----------------
__global__ void gin_ln_reduce(const float* __restrict__ h, float* __restrict__ part, int M) {
    __shared__ float ss[256], sq[256];
    float s = 0.0f, q = 0.0f;
    for (long long i = (long long)blockIdx.x * 256 + threadIdx.x; i < M;
         i += (long long)gridDim.x * 256) {
        float v = h[i];
        s += v;
        q += v * v;
    }
    ss[threadIdx.x] = s;
    sq[threadIdx.x] = q;
    __syncthreads();
    for (int o = 128; o; o >>= 1) {
        if (threadIdx.x < o) {
            ss[threadIdx.x] += ss[threadIdx.x + o];
            sq[threadIdx.x] += sq[threadIdx.x + o];
        }
        __syncthreads();
    }
    if (threadIdx.x == 0) {
        part[blockIdx.x] = ss[0];
        part[gridDim.x + blockIdx.x] = sq[0];
    }
}

__global__ void gin_ln_finalize(const float* __restrict__ part, int nb,
                                float* __restrict__ stats, float invM) {
    __shared__ float ss[256], sq[256];
    float s = 0.0f, q = 0.0f;
    for (int i = threadIdx.x; i < nb; i += 256) {
        s += part[i];
        q += part[nb + i];
    }
    ss[threadIdx.x] = s;
    sq[threadIdx.x] = q;
    __syncthreads();
    for (int o = 128; o; o >>= 1) {
        if (threadIdx.x < o) {
            ss[threadIdx.x] += ss[threadIdx.x + o];
            sq[threadIdx.x] += sq[threadIdx.x + o];
        }
        __syncthreads();
    }
    if (threadIdx.x == 0) {
        float mean = ss[0] * invM;
        float var  = sq[0] * invM - mean * mean;
        var = var > 0.0f ? var : 0.0f;
        stats[0] = mean;
        stats[1] = 1.0f / (sqrtf(var) + LN_EPSF);  // matches ref: /(sqrt(var)+eps)
    }
}

__global__ void gin_ln_apply_relu(const float* __restrict__ in,
                                  const float* __restrict__ stats,
                                  const float* __restrict__ w, const float* __restrict__ b,
                                  float* __restrict__ out, int M) {
    int i = blockIdx.x * blockDim.x + threadIdx.x;
    if (i < M) {
        float mean = stats[0], inv = stats[1];
        int c = i & 127;
        float v = (in[i] - mean) * inv * w[c] + b[c];
        out[i] = fmaxf(v, 0.0f);
    }
}

// ---------------- final classifier: out[i] = dot(h[i], fcW) + fcb ----------------
// one wave per row, float4 per lane, xor-shuffle reduce
__global__ void gin_final(const float* __restrict__ h, const float* __restrict__ w,
                          const float* __restrict__ b, float* __restrict__ out, int n) {
    int gw = (int)(((long long)blockIdx.x * blockDim.x + threadIdx.x) >> 5);
    int lane = threadIdx.x & 31;
    if (gw >= n) return;  // whole wave uniform
    const float4 hv = *(const float4*)(h + (size_t)gw * HIDF + lane * 4);
    const float4 wv = *(const float4*)(w + lane * 4);
    float s = hv.x * wv.x + hv.y * wv.y + hv.z * wv.z + hv.w * wv.w;
    for (int off = 16; off; off >>= 1) s += __shfl_xor(s, off, 32);
    if (lane == 0) out[gw] = s + b[0];
}

extern "C" void kernel_launch(void* const* d_in, const int* in_sizes, int n_in,
                              void* d_out, int out_size, void* d_ws, size_t ws_size,
                              hipStream_t stream) {
    const float* x   = (const float*)d_in[0];
    const int*   ei  = (const int*)d_in[1];
    const float* W1a = (const float*)d_in[2];
    const float* b1a = (const float*)d_in[3];
    const float* W2a = (const float*)d_in[4];
    const float* b2a = (const float*)d_in[5];
    const float* W1b = (const float*)d_in[6];
    const float* b1b = (const float*)d_in[7];
    const float* W2b = (const float*)d_in[8];
    const float* b2b = (const float*)d_in[9];
    const float* ln1w = (const float*)d_in[10];
    const float* ln1b = (const float*)d_in[11];
    const float* ln2w = (const float*)d_in[12];
    const float* ln2b = (const float*)d_in[13];
    const float* fcW  = (const float*)d_in[14];
    const float* fcb  = (const float*)d_in[15];
    float* out = (float*)d_out;

    const int n = in_sizes[0] / HIDF;  // 50000
    const int e = in_sizes[1] / 2;     // 625000
    const int M = n * HIDF;

    const int gb    = (n + 63) / 64;   // GEMM blocks
    const int n_pad = gb * 64;         // padded rows so GEMM stores are unguarded
    const int Mpad  = n_pad * HIDF;

    float* G     = (float*)d_ws;            // [n_pad,128]
    float* H     = G + (size_t)Mpad;        // [n_pad,128]
    float* part  = H + (size_t)Mpad;        // [2*NB_RED]
    float* stats = part + 2 * NB_RED;       // [2]

    const int zb = (M + 255) / 256;
    const int sb = (int)(((long long)e * 32 + 255) / 256);
    const int fb = (int)(((long long)n * 32 + 255) / 256);
    const float invM = 1.0f / (float)M;

    // ---- GIN layer 1 ----
    gin_zero<<<zb, 256, 0, stream>>>(G, M);
    gin_scatter<<<sb, 256, 0, stream>>>(x, ei, G, e);
    gin_gemm128<true, true><<<gb, 128, 0, stream>>>(x, G, W1a, b1a, H, n);        // H = relu((x+agg)@W1a+b1a)
    gin_gemm128<false, false><<<gb, 128, 0, stream>>>(H, nullptr, W2a, b2a, G, n); // G = H@W2a+b2a
    gin_ln_reduce<<<NB_RED, 256, 0, stream>>>(G, part, M);
    gin_ln_finalize<<<1, 256, 0, stream>>>(part, NB_RED, stats, invM);
    gin_ln_apply_relu<<<zb, 256, 0, stream>>>(G, stats, ln1w, ln1b, H, M);        // H = relu(LN(G))

    // ---- GIN layer 2 ----
    gin_zero<<<zb, 256, 0, stream>>>(G, M);
    gin_scatter<<<sb, 256, 0, stream>>>(H, ei, G, e);
    gin_gemm128<true, true><<<gb, 128, 0, stream>>>(H, G, W1b, b1b, G, n);        // G = relu((H+agg)@W1b+b1b)
    gin_gemm128<false, false><<<gb, 128, 0, stream>>>(G, nullptr, W2b, b2b, H, n); // H = G@W2b+b2b
    gin_ln_reduce<<<NB_RED, 256, 0, stream>>>(H, part, M);
    gin_ln_finalize<<<1, 256, 0, stream>>>(part, NB_RED, stats, invM);
    gin_ln_apply_relu<<<zb, 256, 0, stream>>>(H, stats, ln2w, ln2b, G, M);        // G = relu(LN(H))

    // ---- classifier ----
    gin_final<<<fb, 256, 0, stream>>>(G, fcW, fcb, out, n);
}